// SimpleLatch_55052890800853
// MI455X (gfx1250) — compile-verified
//
#include <hip/hip_runtime.h>

// ---------------------------------------------------------------------------
// SimpleLatch scan, MI455X (gfx1250).
//
//  * one wave32 per batch row; lane = one of 32 segments of 64 timesteps
//  * block = 8 rows (256 threads); the block's x tile (8 x 16 KB, contiguous
//    rows) is staged into LDS with ONE Tensor-Data-Mover descriptor
//    (tensor_load_to_lds), padded 2 dwords / 128 dwords so per-lane segment
//    float2 reads are 8B-aligned ds_load_b64 and LDS-bank-conflict-free.
//  * both sweeps (segment summary + emission) read x from LDS -> x read from
//    DRAM exactly once (64 MiB) + 32 MiB store = roofline-minimal traffic.
//  * activations via gfx1250 v_tanh_f32 (sigmoid(x) = 0.5*tanh(x/2)+0.5),
//    one transcendental per activation instead of exp+rcp.
//  * inter-segment combine: 5-step ds_bpermute scan of affine pairs (a,u).
// ---------------------------------------------------------------------------

namespace {
constexpr int kT        = 2048;
constexpr int kSeg      = 32;              // segments per row == wave lanes
constexpr int kL        = kT / kSeg;       // 64 timesteps per segment
constexpr int kRows     = 8;               // batch rows per block
constexpr int kThreads  = kRows * kSeg;    // 256
constexpr int kRowDw    = kT * 2;          // 4096 dwords of x per row
constexpr int kPadEvery = 128;             // TDM pad interval (dwords)
constexpr int kPadDw    = 2;               // TDM pad amount (dwords)
constexpr int kRowDwPad = kRowDw + (kRowDw / kPadEvery) * kPadDw;  // 4160
constexpr int kSegDwPad = kL * 2 + kPadDw;                         // 130
constexpr int kLdsDw    = kRows * kRowDwPad;   // 33280 dw = 133120 B
}

typedef __attribute__((ext_vector_type(4))) unsigned int u32x4;
typedef __attribute__((ext_vector_type(8))) unsigned int u32x8;

__device__ __forceinline__ float fast_tanh(float x) {
#if __has_builtin(__builtin_amdgcn_tanhf)
  return __builtin_amdgcn_tanhf(x);
#else
  float r;
  asm("v_tanh_f32 %0, %1" : "=v"(r) : "v"(x));
  return r;
#endif
}

__device__ __forceinline__ float fast_sigmoid(float x) {
  return __builtin_fmaf(0.5f, fast_tanh(0.5f * x), 0.5f);
}

// gather value from srcLane (wave32); callers guard invalid lanes
__device__ __forceinline__ float lane_gather(float v, int srcLane) {
  return __int_as_float(
      __builtin_amdgcn_ds_bpermute(srcLane << 2, __float_as_int(v)));
}

__global__ __launch_bounds__(kThreads) void
latch_scan_kernel(const float* __restrict__ x, const float* __restrict__ W,
                  const float* __restrict__ bias, float* __restrict__ out) {
  __shared__ float lds[kLdsDw];

  const int tid   = threadIdx.x;
  const int lane  = tid & 31;
  const int wid   = tid >> 5;                          // row within block
  const long long row0 = (long long)blockIdx.x * kRows;

  // -------- TDM stage: 2D tile = 8 rows x 4096 f32, row stride 4096 --------
  // Issued by wave 0 only (scalar branch: TDM ignores EXEC).
  if (__builtin_amdgcn_readfirstlane(wid) == 0) {
    unsigned long long ga =
        (unsigned long long)(const void*)(x + row0 * kRowDw);
    unsigned lds_off = (unsigned)(unsigned long long)(
        __attribute__((address_space(3))) float*)lds;

    // D# group 0: count=1 | lds_addr | global_addr[56:0] | type=2
    u32x4 g0;
    g0.x = 1u;
    g0.y = lds_off;
    g0.z = (unsigned)ga;
    g0.w = (unsigned)((ga >> 32) & 0x01FFFFFFull) | (2u << 30);

    // D# group 1
    u32x8 g1;
    g1.s0 = (2u << 16)                 // data_size = 4 bytes
          | (1u << 20)                 // pad_enable
          | (6u << 22)                 // pad_interval: 128 dwords
          | (1u << 25);                // pad_amount:   2 dwords
    g1.s1 = ((unsigned)(kRowDw & 0xFFFF)) << 16;        // tensor_dim0 lo16
    g1.s2 = ((unsigned)(kRowDw >> 16) & 0xFFFFu)        // tensor_dim0 hi16
          | (((unsigned)kRows & 0xFFFFu) << 16);        // tensor_dim1 lo16
    g1.s3 = ((unsigned)(kRowDw & 0xFFFF)) << 16;        // tile_dim0 = 4096
    g1.s4 = (unsigned)kRows;                            // tile_dim1=8, dim2=0
    g1.s5 = (unsigned)kRowDw;                           // dim0_stride lo32
    g1.s6 = 0u;                                         // stride hi / dim1_stride
    g1.s7 = 0u;
    asm volatile("tensor_load_to_lds %0, %1" ::"s"(g0), "s"(g1) : "memory");
    __builtin_amdgcn_s_wait_tensorcnt(0);
  }
  __syncthreads();

  // -------- gate parameters (uniform -> scalar loads) --------
  const float w00 = W[0], w01 = W[1];   // forget
  const float w10 = W[2], w11 = W[3];   // input
  const float w20 = W[4], w21 = W[5];   // block
  const float w30 = W[6], w31 = W[7];   // output
  const float b0 = bias[0], b1 = bias[1], b2 = bias[2], b3 = bias[3];

  // Segment = exactly one 128-dword pad interval -> pad offset is constant
  // per segment; base dword = wid*4160 + lane*130 (even -> 8B aligned).
  const float2* __restrict__ seg =
      (const float2*)(lds + wid * kRowDwPad + lane * kSegDwPad);

  // -------- sweep 1: per-segment affine summary c_out = a*c_in + u --------
  float a = 1.0f, u = 0.0f;
#pragma unroll 8
  for (int k = 0; k < kL; ++k) {
    const float2 xv = seg[k];
    const float f = fast_sigmoid(__builtin_fmaf(xv.x, w00, __builtin_fmaf(xv.y, w01, b0)));
    const float i = fast_sigmoid(__builtin_fmaf(xv.x, w10, __builtin_fmaf(xv.y, w11, b1)));
    const float g = fast_tanh   (__builtin_fmaf(xv.x, w20, __builtin_fmaf(xv.y, w21, b2)));
    u = __builtin_fmaf(u, f, i * g);
    a *= f;
  }

  // -------- wave-level inclusive scan of affine pairs (Hillis-Steele) -----
  // compose(up, mine): A = A_m*A_u ; U = A_m*U_u + U_m
  float A = a, U = u;
#pragma unroll
  for (int d = 1; d < kSeg; d <<= 1) {
    const float Au = lane_gather(A, lane - d);
    const float Uu = lane_gather(U, lane - d);
    if (lane >= d) {
      U = __builtin_fmaf(A, Uu, U);
      A = A * Au;
    }
  }
  const float cprev = lane_gather(U, lane - 1);  // exclusive prefix (c0 = 0)
  float c = (lane == 0) ? 0.0f : cprev;

  // -------- sweep 2: replay segment with correct initial c, emit y --------
  float* __restrict__ orow = out + (row0 + wid) * (long long)kT + lane * kL;
#pragma unroll 8
  for (int k = 0; k < kL; ++k) {
    const float2 xv = seg[k];
    const float f = fast_sigmoid(__builtin_fmaf(xv.x, w00, __builtin_fmaf(xv.y, w01, b0)));
    const float i = fast_sigmoid(__builtin_fmaf(xv.x, w10, __builtin_fmaf(xv.y, w11, b1)));
    const float g = fast_tanh   (__builtin_fmaf(xv.x, w20, __builtin_fmaf(xv.y, w21, b2)));
    const float o = fast_sigmoid(__builtin_fmaf(xv.x, w30, __builtin_fmaf(xv.y, w31, b3)));
    c = __builtin_fmaf(c, f, i * g);
    orow[k] = o * fast_tanh(c);
  }
}

extern "C" void kernel_launch(void* const* d_in, const int* in_sizes, int n_in,
                              void* d_out, int out_size, void* d_ws, size_t ws_size,
                              hipStream_t stream) {
  (void)n_in; (void)out_size; (void)d_ws; (void)ws_size;
  const float* x    = (const float*)d_in[0];   // [B, T, 2]
  const float* W    = (const float*)d_in[1];   // [4, 2]
  const float* bias = (const float*)d_in[2];   // [4]
  float* out        = (float*)d_out;           // [B, T, 1]

  const int B = in_sizes[0] / (2 * kT);        // 4096
  dim3 grid(B / kRows);
  dim3 block(kThreads);
  latch_scan_kernel<<<grid, block, 0, stream>>>(x, W, bias, out);
}